// RationalQuadraticSpline_51754355917073
// MI455X (gfx1250) — compile-verified
//
#include <hip/hip_runtime.h>
#include <hip/hip_bf16.h>

#define NB      30
#define MINBW   0.001f
#define MINBH   0.001f
#define MIND    0.001f
#define EPSF    1e-6f

#define COLS    32      // columns (variables) per block tile
#define ROWSPB  128     // rows per block
#define TPB     256     // threads per block

typedef int v4i __attribute__((ext_vector_type(4)));

// ---------------- gfx1250 async-to-LDS path (guarded) ----------------
#if defined(__has_builtin)
#  if __has_builtin(__builtin_amdgcn_global_load_async_to_lds_b128)
#    define USE_ASYNC_LDS 1
#  endif
#  if __has_builtin(__builtin_amdgcn_s_wait_asynccnt)
#    define USE_WAIT_BUILTIN 1
#  endif
#endif

__device__ __forceinline__ void copy16_to_lds(const float* g, float* l) {
#ifdef USE_ASYNC_LDS
  // signature: (v4i addrspace(1)*, v4i addrspace(3)*, imm offset, imm cpol)
  __builtin_amdgcn_global_load_async_to_lds_b128(
      (__attribute__((address_space(1))) v4i*)g,
      (__attribute__((address_space(3))) v4i*)l,
      0, 0);
#else
  *(float4*)l = *(const float4*)g;
#endif
}

__device__ __forceinline__ void wait_async_copies() {
#ifdef USE_ASYNC_LDS
#  ifdef USE_WAIT_BUILTIN
  __builtin_amdgcn_s_wait_asynccnt(0);
#  else
  asm volatile("s_wait_asynccnt 0x0" ::: "memory");
#  endif
#endif
}

// ---------------- helpers ----------------
__device__ __forceinline__ float softplusf(float x) {
  return (x > 20.0f) ? x : log1pf(__expf(x));
}

// ---------------- kernel A: expand per-variable spline tables ----------------
// ws_cb : [V][32]  bin boundaries; [30] = last + EPS, [31] = +inf
// ws_tab: [V][NB][8] = {cw, w, ch, h, delta, d0, d1, 0}
__global__ void rqs_prep_kernel(const float* __restrict__ uw,
                                const float* __restrict__ uh,
                                const float* __restrict__ ud,
                                float* __restrict__ ws_cb,
                                float* __restrict__ ws_tab,
                                int V) {
  int v = blockIdx.x * blockDim.x + threadIdx.x;
  if (v >= V) return;

  float cw[NB + 1], ch[NB + 1], dv[NB + 1], t[NB];

  // widths: softmax -> floor -> cumsum (LI=0, UI=1 so affine is identity)
  {
    const float* p = uw + v * NB;
    float m = -3.4e38f;
#pragma unroll
    for (int k = 0; k < NB; ++k) { t[k] = p[k]; m = fmaxf(m, t[k]); }
    float s = 0.0f;
#pragma unroll
    for (int k = 0; k < NB; ++k) { t[k] = __expf(t[k] - m); s += t[k]; }
    float scale = (1.0f - MINBW * NB) / s;
    float run = 0.0f; cw[0] = 0.0f;
#pragma unroll
    for (int k = 0; k < NB; ++k) { run += MINBW + scale * t[k]; cw[k + 1] = run; }
  }
  // heights
  {
    const float* p = uh + v * NB;
    float m = -3.4e38f;
#pragma unroll
    for (int k = 0; k < NB; ++k) { t[k] = p[k]; m = fmaxf(m, t[k]); }
    float s = 0.0f;
#pragma unroll
    for (int k = 0; k < NB; ++k) { t[k] = __expf(t[k] - m); s += t[k]; }
    float scale = (1.0f - MINBH * NB) / s;
    float run = 0.0f; ch[0] = 0.0f;
#pragma unroll
    for (int k = 0; k < NB; ++k) { run += MINBH + scale * t[k]; ch[k + 1] = run; }
  }
  // derivatives: edges pinned so deriv == 1 at boundaries
  {
    const float cc = logf(expf(1.0f - MIND) - 1.0f);
    float edge = MIND + softplusf(cc);
    dv[0] = edge; dv[NB] = edge;
    const float* p = ud + v * (NB - 1);
#pragma unroll
    for (int k = 1; k < NB; ++k) dv[k] = MIND + softplusf(p[k - 1]);
  }

  float* cb = ws_cb + (size_t)v * 32;
#pragma unroll
  for (int k = 0; k < NB; ++k) cb[k] = cw[k];
  cb[NB]     = cw[NB] + EPSF;
  cb[NB + 1] = __int_as_float(0x7f800000);  // +inf pad for 5-step search

  float* tab = ws_tab + (size_t)v * NB * 8;
#pragma unroll
  for (int k = 0; k < NB; ++k) {
    float w = cw[k + 1] - cw[k];
    float h = ch[k + 1] - ch[k];
    tab[k * 8 + 0] = cw[k];
    tab[k * 8 + 1] = w;
    tab[k * 8 + 2] = ch[k];
    tab[k * 8 + 3] = h;
    tab[k * 8 + 4] = h / w;     // delta
    tab[k * 8 + 5] = dv[k];     // d0
    tab[k * 8 + 6] = dv[k + 1]; // d1
    tab[k * 8 + 7] = 0.0f;
  }
}

// ---------------- kernel B: streaming spline evaluation ----------------
__global__ __launch_bounds__(TPB) void rqs_main_kernel(
    const float* __restrict__ in,
    const float* __restrict__ ws_cb,
    const float* __restrict__ ws_tab,
    float* __restrict__ out,
    float* __restrict__ lad,
    int B, int V) {
  __shared__ __attribute__((aligned(16))) float s_cb[COLS * 32];        // 4 KB
  __shared__ __attribute__((aligned(16))) float s_tab[COLS * NB * 8];   // 30 KB

  const int tid = threadIdx.x;
  const int v0  = blockIdx.x * COLS;
  const int r0  = blockIdx.y * ROWSPB;

  // Async-copy this block's table slice into LDS (contiguous in ws).
  {
    const float* g = ws_cb + (size_t)v0 * 32;
    for (int i = tid; i < (COLS * 32) / 4; i += TPB)
      copy16_to_lds(g + i * 4, s_cb + i * 4);
    const float* g2 = ws_tab + (size_t)v0 * NB * 8;
    for (int i = tid; i < (COLS * NB * 8) / 4; i += TPB)
      copy16_to_lds(g2 + i * 4, s_tab + i * 4);
  }
  wait_async_copies();
  __syncthreads();

  const int col = tid & (COLS - 1);       // lane -> consecutive v (coalesced)
  const int rof = tid >> 5;               // 8 row groups per block (wave-uniform)
  const float* cbp = s_cb + col * 32;

  for (int r = rof; r < ROWSPB; r += TPB / COLS) {
    const int row = r0 + r;
    if (row >= B) break;                  // wave-uniform
    const int idx = row * V + v0 + col;

    const float x = in[idx];
    __builtin_prefetch(in + idx + (TPB / COLS) * 2 * V, 0, 3);  // stream ahead

    const float xc = fminf(fmaxf(x, 0.0f), 1.0f);

    // branchless 5-step binary search over 32 padded boundaries -> bin in [0,29]
    int b = 0;
    b += (cbp[b + 16] <= xc) ? 16 : 0;
    b += (cbp[b +  8] <= xc) ?  8 : 0;
    b += (cbp[b +  4] <= xc) ?  4 : 0;
    b += (cbp[b +  2] <= xc) ?  2 : 0;
    b += (cbp[b +  1] <= xc) ?  1 : 0;

    const float* sp = s_tab + (col * NB + b) * 8;
    const float4 A  = *(const float4*)(sp);      // cw, w, ch, h
    const float4 Bv = *(const float4*)(sp + 4);  // delta, d0, d1, pad

    const float theta = (xc - A.x) / A.y;
    const float omt   = 1.0f - theta;
    const float t1mt  = theta * omt;
    const float den   = Bv.x + (Bv.y + Bv.z - 2.0f * Bv.x) * t1mt;
    const float o_in  = A.z + A.w * (Bv.x * theta * theta + Bv.y * t1mt) / den;
    const float numd  = (Bv.x * Bv.x) *
                        (Bv.z * theta * theta + 2.0f * Bv.x * t1mt + Bv.y * omt * omt);
    const float l_in  = __logf(numd) - 2.0f * __logf(den);

    const bool inside = !(x < 0.0f) && !(x > 1.0f);
    // outside: identity tails (LO=LI, UO=UI, DERIV_OUT=1) -> out = x, lad = 0
    out[idx] = inside ? o_in : x;
    lad[idx] = inside ? l_in : 0.0f;
  }
}

// ---------------- launch ----------------
extern "C" void kernel_launch(void* const* d_in, const int* in_sizes, int n_in,
                              void* d_out, int out_size, void* d_ws, size_t ws_size,
                              hipStream_t stream) {
  const float* in = (const float*)d_in[0];
  const float* uw = (const float*)d_in[1];
  const float* uh = (const float*)d_in[2];
  const float* ud = (const float*)d_in[3];

  const int V = in_sizes[1] / NB;        // 512
  const int B = in_sizes[0] / V;         // 16384

  float* ws_cb  = (float*)d_ws;                    // V*32 floats
  float* ws_tab = ws_cb + (size_t)V * 32;          // V*NB*8 floats

  float* out = (float*)d_out;
  float* lad = out + (size_t)B * V;

  rqs_prep_kernel<<<(V + 127) / 128, 128, 0, stream>>>(uw, uh, ud, ws_cb, ws_tab, V);

  dim3 grid(V / COLS, (B + ROWSPB - 1) / ROWSPB);  // 16 x 128 = 2048 blocks
  rqs_main_kernel<<<grid, TPB, 0, stream>>>(in, ws_cb, ws_tab, out, lad, B, V);
}